// VectorQuantizationLayer_16776142258550
// MI455X (gfx1250) — compile-verified
//
#include <hip/hip_runtime.h>

typedef __attribute__((ext_vector_type(2))) float v2f;
typedef __attribute__((ext_vector_type(8))) float v8f;

#define DD 64           // vector dimension
#define KK 1024         // codebook size
#define KC 128          // codebook columns per LDS chunk
#define NCHUNK (KK / KC)
#define LDS_STRIDE 68   // 64 + 4 floats padding -> conflict-free b64 fragment loads
#define WAVES 8
#define ROWS_PER_WG 256 // 8 waves x 2 row-tiles x 16 rows
#define NROWS 131072    // 32*64*64

__global__ __launch_bounds__(256)
void vq_main_kernel(const float* __restrict__ X, const float* __restrict__ E,
                    float* __restrict__ out, float* __restrict__ partial) {
    __shared__ float ldsB[KC * LDS_STRIDE];
    __shared__ float ldsE2[KC];
    __shared__ float ldsPart[WAVES];

    const int tid  = threadIdx.x;
    const int wave = tid >> 5;
    const int lane = tid & 31;
    const int lh   = lane & 15;   // column / row-in-tile lane id
    const int h    = lane >> 4;   // half-wave id

    const int rowBase = blockIdx.x * ROWS_PER_WG + wave * 32;
    const int r0 = rowBase;       // tile 0: rows r0..r0+15
    const int r1 = rowBase + 16;  // tile 1: rows r1..r1+15

    // ---- A fragments in registers for the whole kernel ----
    // lane l holds row (l&15); k-step s supplies K = 4s + {0,1} + 2h  -> one float2
    v2f a0[16], a1[16];
    {
        const float* x0 = X + (size_t)(r0 + lh) * DD + 2 * h;
        const float* x1 = X + (size_t)(r1 + lh) * DD + 2 * h;
#pragma unroll
        for (int s = 0; s < 16; ++s) {
            a0[s] = *(const v2f*)(x0 + 4 * s);
            a1[s] = *(const v2f*)(x1 + 4 * s);
        }
    }

    // running argmax of (sim - 0.5*||e||^2)  ==  argmin of the reference distance
    float best0[8], best1[8];
    int   bi0[8],   bi1[8];
#pragma unroll
    for (int j = 0; j < 8; ++j) {
        best0[j] = -3.402823466e38f; best1[j] = -3.402823466e38f;
        bi0[j] = 0; bi1[j] = 0;
    }

    for (int c = 0; c < NCHUNK; ++c) {
        __syncthreads();
        // Stage E chunk transposed: ldsB[k*STRIDE + d] = E[d*KK + c*KC + k]
        for (int i = tid; i < DD * KC; i += 256) {
            int d  = i >> 7;        // i / KC
            int kk = i & (KC - 1);  // i % KC
            ldsB[kk * LDS_STRIDE + d] = E[(size_t)d * KK + c * KC + kk];
        }
        __syncthreads();
        if (tid < KC) {
            float s = 0.f;
#pragma unroll
            for (int d = 0; d < DD; ++d) {
                float v = ldsB[tid * LDS_STRIDE + d];
                s += v * v;
            }
            ldsE2[tid] = 0.5f * s;
        }
        __syncthreads();

#pragma unroll 2
        for (int t = 0; t < KC / 16; ++t) {
            // prefetch the whole B tile into registers first so the
            // ds_load latency is hidden under the WMMA chain
            const float* bp = &ldsB[(t * 16 + lh) * LDS_STRIDE + 2 * h];
            v2f bf[16];
#pragma unroll
            for (int s = 0; s < 16; ++s) {
                bf[s] = *(const v2f*)(bp + 4 * s);   // B frag: K = 4s+{0,1}+2h, N = lh
            }
            const float e2h = ldsE2[t * 16 + lh];

            v8f c0 = {}; v8f c1 = {};
#pragma unroll
            for (int s = 0; s < 16; ++s) {
                c0 = __builtin_amdgcn_wmma_f32_16x16x4_f32(
                        false, a0[s], false, bf[s], (short)0, c0, false, false);
                c1 = __builtin_amdgcn_wmma_f32_16x16x4_f32(
                        false, a1[s], false, bf[s], (short)0, c1, false, false);
            }

            const int col = c * KC + t * 16 + lh;
#pragma unroll
            for (int j = 0; j < 8; ++j) {           // C: row = j + 8h, col = lh
                float s0 = c0[j] - e2h;
                if (s0 > best0[j]) { best0[j] = s0; bi0[j] = col; }
                float s1 = c1[j] - e2h;
                if (s1 > best1[j]) { best1[j] = s1; bi1[j] = col; }
            }
        }
    }

    // ---- reduce argmax across the 16 lanes of each half (tie -> smaller index) ----
#pragma unroll
    for (int j = 0; j < 8; ++j) {
#pragma unroll
        for (int m = 8; m >= 1; m >>= 1) {
            float ov = __shfl_xor(best0[j], m);
            int   oi = __shfl_xor(bi0[j],   m);
            if (ov > best0[j] || (ov == best0[j] && oi < bi0[j])) { best0[j] = ov; bi0[j] = oi; }
            float ov1 = __shfl_xor(best1[j], m);
            int   oi1 = __shfl_xor(bi1[j],   m);
            if (ov1 > best1[j] || (ov1 == best1[j] && oi1 < bi1[j])) { best1[j] = ov1; bi1[j] = oi1; }
        }
    }

    // ---- gather codebook rows, write output, accumulate squared error ----
    float err = 0.f;
#pragma unroll
    for (int j = 0; j < 8; ++j) {
        const int rowA = r0 + j + 8 * h;  const int kA = bi0[j];
        const int rowB = r1 + j + 8 * h;  const int kB = bi1[j];
#pragma unroll
        for (int ddp = 0; ddp < 4; ++ddp) {
            const int d = lh + 16 * ddp;
            float qA = E[(size_t)d * KK + kA];
            float xA = X[(size_t)rowA * DD + d];
            out[(size_t)rowA * DD + d] = qA;
            float fA = qA - xA; err += fA * fA;
            float qB = E[(size_t)d * KK + kB];
            float xB = X[(size_t)rowB * DD + d];
            out[(size_t)rowB * DD + d] = qB;
            float fB = qB - xB; err += fB * fB;
        }
    }
#pragma unroll
    for (int m = 1; m < 32; m <<= 1) err += __shfl_xor(err, m);
    if (lane == 0) ldsPart[wave] = err;
    __syncthreads();
    if (tid == 0) {
        float s = 0.f;
#pragma unroll
        for (int w = 0; w < WAVES; ++w) s += ldsPart[w];
        partial[blockIdx.x] = s;
    }
}

__global__ void vq_loss_kernel(const float* __restrict__ partial, int n,
                               float* __restrict__ out_loss) {
    if (blockIdx.x == 0 && threadIdx.x == 0) {
        float s = 0.f;
        for (int i = 0; i < n; ++i) s += partial[i];   // fixed order -> deterministic
        const float mean = s / (float)((size_t)NROWS * DD);
        out_loss[0] = 1.25f * mean;                    // BETA*mse + mse, BETA = 0.25
    }
}

extern "C" void kernel_launch(void* const* d_in, const int* in_sizes, int n_in,
                              void* d_out, int out_size, void* d_ws, size_t ws_size,
                              hipStream_t stream) {
    const float* X = (const float*)d_in[0];   // [32,64,64,64] f32
    const float* E = (const float*)d_in[1];   // [64,1024]     f32
    float* out     = (float*)d_out;           // [N*64] quantized_ste, then loss scalar
    float* partial = (float*)d_ws;            // 512 floats of scratch

    const int nWG = NROWS / ROWS_PER_WG;      // 512
    vq_main_kernel<<<dim3(nWG), dim3(256), 0, stream>>>(X, E, out, partial);
    vq_loss_kernel<<<dim3(1), dim3(1), 0, stream>>>(partial, nWG, out + (size_t)NROWS * DD);
}